// GRUModel_47132971107185
// MI455X (gfx1250) — compile-verified
//
#include <hip/hip_runtime.h>

// ---------------------------------------------------------------------------
// Fused GRU for MI455X (gfx1250, wave32).
//
// S=1024, B=4096, H=8, 3H=24.  Memory-bound (~262 MB min traffic, ~11us at
// 23.3 TB/s), but the recurrence is latency-bound per wave, so the design
// minimizes the serial per-step critical path through h.
//
// TRANSPOSED formulation: instead of h[16x8] @ W^T[8x24], compute
// W[24x8] @ h^T[8x16] with V_WMMA_F32_16X16X4_F32:
//   A = weights (STATIC, in registers for the whole kernel):
//       A 16x4 layout: lane l holds row (l%16), K pair 2*(l/16)
//       2 M-tiles: gate rows 0..15 (r|z) and 16..31 (n | zero padding)
//   B = h^T (8x16): lane l holds B[k][n] = h[batch c][k], c = l%16,
//       k pair = 2*(l/16) (+4 for the second K chunk)
//   D (16x16): lane l, VGPR v -> gate row v + 8*(l/16), batch col c
//
// Payoff: lane c's 8 D-tile VGPRs are exactly the gates of batch element c,
// and after the gate fusion lane c holds the complete NEW h row h[c][0..7]
// in registers.  The next step's B operand needs only 4 ds_swizzle(xor 16)
// + v_cndmask (lanes 16..31 fetch h[c][2,3,6,7] from lane c) — the LDS
// store->wait->load transpose round-trip of the previous version is gone
// (no LDS memory traffic at all).
//
// Gate combine: r = D0 rows 0..7 (lanes 0..15), z = D0 rows 8..15
// (lanes 16..31), n-pre = D1 rows 16..23 -> one xor-16 ds_swizzle per
// accumulator VGPR delivers z to the r lanes.  Weight rows 24..31 are
// zero-padded so dead-lane math stays finite (tanh(0)=0); garbage never
// crosses back through the swizzle selects.
//
// Nonlinearities on the fast TRANS path (round-1 fix):
//   sigmoid = v_rcp(1+v_exp(-x)), tanh = 1-2*v_rcp(1+v_exp(2x)).
// ---------------------------------------------------------------------------

typedef __attribute__((ext_vector_type(2))) float v2f;
typedef __attribute__((ext_vector_type(4))) float v4f;
typedef __attribute__((ext_vector_type(8))) float v8f;

#define S_LEN 1024
#define B_SZ  4096
#define H_SZ  8
#define G_SZ  24  // 3*H

// ds_swizzle group-of-32: offset = xor<<10 | or<<5 | and ; xor=16, and=0x1f
__device__ __forceinline__ float swz_xor16(float x) {
  return __int_as_float(
      __builtin_amdgcn_ds_swizzle(__float_as_int(x), 0x401f));
}

__device__ __forceinline__ float fast_sigmoid(float x) {
  return __builtin_amdgcn_rcpf(1.0f + __expf(-x));  // v_exp + v_rcp
}

__device__ __forceinline__ float fast_tanh(float x) {
  return 1.0f - 2.0f * __builtin_amdgcn_rcpf(1.0f + __expf(2.0f * x));
}

__device__ __forceinline__ v8f wmma_f32(v2f a, v2f b, v8f c) {
  // 8 args: (neg_a, A, neg_b, B, c_mod, C, reuse_a, reuse_b)
  return __builtin_amdgcn_wmma_f32_16x16x4_f32(false, a, false, b, (short)0, c,
                                               false, false);
}

__global__ __launch_bounds__(32) void gru_wmma_kernel(
    const float* __restrict__ input_,  // (S, B, H)
    const float* __restrict__ w_ih,    // (3H, H) row-major
    const float* __restrict__ w_hh,    // (3H, H)
    const float* __restrict__ b_ih,    // (3H)
    const float* __restrict__ b_hh,    // (3H)
    float* __restrict__ out) {         // (S*B*H) outputs ++ (B*H) h_last
  const int lane = threadIdx.x;        // 0..31
  const int c    = lane & 15;          // batch column within tile / A row
  const int part = lane >> 4;          // 0 or 1
  const int b0   = blockIdx.x * 16;    // batch tile origin
  const int k0   = 2 * part;           // K pair held by this lane

  // ---- weights as STATIC WMMA A operands: row = mt*16+c, k = kk*4+k0 ------
  v2f Aih[2][2], Ahh[2][2];
#pragma unroll
  for (int mt = 0; mt < 2; ++mt) {
#pragma unroll
    for (int kk = 0; kk < 2; ++kk) {
      const int g = mt * 16 + c;   // gate row
      const int k = kk * 4 + k0;
      v2f wi = {0.0f, 0.0f}, wh = {0.0f, 0.0f};
      if (g < G_SZ) {
        wi.x = w_ih[g * H_SZ + k];
        wi.y = w_ih[g * H_SZ + k + 1];
        wh.x = w_hh[g * H_SZ + k];
        wh.y = w_hh[g * H_SZ + k + 1];
      }
      Aih[mt][kk] = wi;
      Ahh[mt][kk] = wh;
    }
  }

  // ---- biases as WMMA C initializers: C[g][n] = bias[g], g = v + 8*part ---
  v8f cbx0, cbh0, cbx1, cbh1;
#pragma unroll
  for (int v = 0; v < 8; ++v) {
    cbx0[v] = b_ih[v + 8 * part];                  // rows 0..15  (r|z)
    cbh0[v] = b_hh[v + 8 * part];
    cbx1[v] = part ? 0.0f : b_ih[16 + v];          // rows 16..23 (n), pad 0
    cbh1[v] = part ? 0.0f : b_hh[16 + v];
  }

  // ---- h0 = 0: lane c (part 0) owns row h[c][0..7]; B operand in regs -----
  float hD[8];
#pragma unroll
  for (int v = 0; v < 8; ++v) hD[v] = 0.0f;
  v2f hB0 = {0.0f, 0.0f};  // B for K chunk 0..3 : {h[c][k0],   h[c][k0+1]}
  v2f hB1 = {0.0f, 0.0f};  // B for K chunk 4..7 : {h[c][4+k0], h[c][5+k0]}

  for (int t = 0; t < S_LEN; ++t) {
    const float* xt = input_ + ((size_t)t * B_SZ + b0) * H_SZ;

    // x^T B operand straight from global: lane -> batch row c, K pair k0
    v2f xB0 = *(const v2f*)(xt + c * H_SZ + k0);
    v2f xB1 = *(const v2f*)(xt + c * H_SZ + k0 + 4);
    if (t + 1 < S_LEN)  // warm caches for next step (global_prefetch_b8)
      __builtin_prefetch(xt + (size_t)B_SZ * H_SZ + c * H_SZ, 0, 3);

    // ---- 8x V_WMMA_F32_16X16X4_F32 (A = static weights) ------------------
    v8f cx0 = cbx0, cx1 = cbx1, ch0 = cbh0, ch1 = cbh1;
    cx0 = wmma_f32(Aih[0][0], xB0, cx0);
    cx0 = wmma_f32(Aih[0][1], xB1, cx0);
    cx1 = wmma_f32(Aih[1][0], xB0, cx1);
    cx1 = wmma_f32(Aih[1][1], xB1, cx1);
    ch0 = wmma_f32(Ahh[0][0], hB0, ch0);
    ch0 = wmma_f32(Ahh[0][1], hB1, ch0);
    ch1 = wmma_f32(Ahh[1][0], hB0, ch1);
    ch1 = wmma_f32(Ahh[1][1], hB1, ch1);

    // ---- fused gates: r=sig(xr+hr), z=sig(xz+hz), n=tanh(xn + r*hn) ------
    // lanes 0..15: s0 = r[gate v]; lanes 16..31: s0 = z[gate v]
#pragma unroll
    for (int v = 0; v < 8; ++v) {
      float s0 = fast_sigmoid(cx0[v] + ch0[v]);
      float z  = swz_xor16(s0);                   // r-lanes receive their z
      float n  = fast_tanh(cx1[v] + s0 * ch1[v]); // valid in r-lanes
      hD[v]    = n + z * (hD[v] - n);             // (1-z)*n + z*h
    }

    // ---- rebuild h^T B operand: lanes 16..31 fetch k=2,3,6,7 from lane c -
    {
      float s2 = swz_xor16(hD[2]);
      float s3 = swz_xor16(hD[3]);
      float s6 = swz_xor16(hD[6]);
      float s7 = swz_xor16(hD[7]);
      hB0 = part ? (v2f){s2, s3} : (v2f){hD[0], hD[1]};
      hB1 = part ? (v2f){s6, s7} : (v2f){hD[4], hD[5]};
    }

    // ---- output store: lane c owns row h[c][0..7] -> 2 half-wave b128 ----
    if (part == 0) {
      float* op = out + ((size_t)t * B_SZ + b0 + c) * H_SZ;
      *(v4f*)(op)     = (v4f){hD[0], hD[1], hD[2], hD[3]};
      *(v4f*)(op + 4) = (v4f){hD[4], hD[5], hD[6], hD[7]};
    }
  }

  // ---- h_last (1, B, H) appended after outputs ----------------------------
  if (part == 0) {
    float* op = out + (size_t)S_LEN * B_SZ * H_SZ + ((size_t)b0 + c) * H_SZ;
    *(v4f*)(op)     = (v4f){hD[0], hD[1], hD[2], hD[3]};
    *(v4f*)(op + 4) = (v4f){hD[4], hD[5], hD[6], hD[7]};
  }
}

extern "C" void kernel_launch(void* const* d_in, const int* in_sizes, int n_in,
                              void* d_out, int out_size, void* d_ws,
                              size_t ws_size, hipStream_t stream) {
  (void)in_sizes; (void)n_in; (void)out_size; (void)d_ws; (void)ws_size;
  const float* input_ = (const float*)d_in[0];
  const float* w_ih   = (const float*)d_in[1];
  const float* w_hh   = (const float*)d_in[2];
  const float* b_ih   = (const float*)d_in[3];
  const float* b_hh   = (const float*)d_in[4];
  float* out = (float*)d_out;

  // 4096 / 16 = 256 independent waves, one per workgroup.
  gru_wmma_kernel<<<B_SZ / 16, 32, 0, stream>>>(input_, w_ih, w_hh, b_ih,
                                                b_hh, out);
}